// GraphMET_GCNConv_59021440582021
// MI455X (gfx1250) — compile-verified
//
#include <hip/hip_runtime.h>
#include <hip/hip_bf16.h>

#define N_NODES 100000
#define N_EDGES 1200000
#define HID 64
#define BN_EPS 1e-5f

typedef float v2f __attribute__((ext_vector_type(2)));
typedef float v8f __attribute__((ext_vector_type(8)));

__device__ __forceinline__ float elu_f(float v) { return v > 0.f ? v : expm1f(v); }

// ---------------------------------------------------------------------------
// Degree histogram over destinations (int atomics; needed for dinv and CSR).
// ---------------------------------------------------------------------------
__global__ __launch_bounds__(256) void deg_kernel(const int* __restrict__ ei,
                                                  int* __restrict__ deg) {
  int e = blockIdx.x * blockDim.x + threadIdx.x;
  if (e >= N_EDGES) return;
  atomicAdd(&deg[ei[N_EDGES + e]], 1);
}

__global__ __launch_bounds__(256) void dinv_kernel(const int* __restrict__ deg,
                                                   float* __restrict__ dinv) {
  int i = blockIdx.x * blockDim.x + threadIdx.x;
  if (i >= N_NODES) return;
  dinv[i] = rsqrtf((float)deg[i] + 1.0f);
}

// ---------------------------------------------------------------------------
// Single-block exclusive scan of deg -> row_ptr (and a copy into nxt for the
// atomic bucket fill). 100K elements, 1024-thread LDS Hillis-Steele chunks.
// ---------------------------------------------------------------------------
__global__ __launch_bounds__(1024) void scan_kernel(const int* __restrict__ deg,
                                                    int* __restrict__ row_ptr,
                                                    int* __restrict__ nxt, int n) {
  __shared__ int s[1024];
  __shared__ int carry_s;
  int t = threadIdx.x;
  if (t == 0) carry_s = 0;
  __syncthreads();
  for (int base = 0; base < n; base += 1024) {
    int i = base + t;
    int v = (i < n) ? deg[i] : 0;
    s[t] = v;
    __syncthreads();
    for (int offd = 1; offd < 1024; offd <<= 1) {
      int tmp = (t >= offd) ? s[t - offd] : 0;
      __syncthreads();
      s[t] += tmp;
      __syncthreads();
    }
    int carry = carry_s;
    int excl = carry + s[t] - v;
    if (i < n) { row_ptr[i] = excl; nxt[i] = excl; }
    __syncthreads();
    if (t == 0) carry_s = carry + s[1023];
    __syncthreads();
  }
  if (t == 0) row_ptr[n] = carry_s;
}

__global__ __launch_bounds__(256) void fill_kernel(const int* __restrict__ ei,
                                                   int* __restrict__ nxt,
                                                   int* __restrict__ col) {
  int e = blockIdx.x * blockDim.x + threadIdx.x;
  if (e >= N_EDGES) return;
  int d = ei[N_EDGES + e];
  int p = atomicAdd(&nxt[d], 1);
  col[p] = ei[e];
}

// ---------------------------------------------------------------------------
// Front end: per-node continuous MLP (8->32), embedding lookups, cat MLP
// (48->32), concat -> z (N x 64). Weights are uniform -> scalar loads.
// ---------------------------------------------------------------------------
__global__ __launch_bounds__(256) void frontend_kernel(
    const float* __restrict__ x, const float* __restrict__ ecw,
    const float* __restrict__ epw, const float* __restrict__ efw,
    const float* __restrict__ w_cont, const float* __restrict__ b_cont,
    const float* __restrict__ w_cat, const float* __restrict__ b_cat,
    float* __restrict__ z) {
  int i = blockIdx.x * blockDim.x + threadIdx.x;
  if (i >= N_NODES) return;
  const float* xr = x + (size_t)i * 11;
  float xc[8];
#pragma unroll
  for (int k = 0; k < 8; ++k) xc[k] = xr[k];
  int pdg = (int)xr[8];
  int charge = (int)xr[9];
  int fpv = (int)xr[10];
  int p = pdg < 0 ? -pdg : pdg;
  int pidx;
  switch (p) {
    case 1:   pidx = 0; break;
    case 2:   pidx = 1; break;
    case 11:  pidx = 2; break;
    case 13:  pidx = 3; break;
    case 22:  pidx = 4; break;
    case 130: pidx = 5; break;
    default:  pidx = 6; break;  // 211
  }
  float cat[48];
#pragma unroll
  for (int j = 0; j < 16; ++j) {
    cat[j]      = ecw[(charge + 1) * 16 + j];
    cat[16 + j] = epw[pidx * 16 + j];
    cat[32 + j] = efw[fpv * 16 + j];
  }
  float* zr = z + (size_t)i * 64;
  for (int j = 0; j < 32; ++j) {
    float acc = b_cat[j];
#pragma unroll
    for (int k = 0; k < 48; ++k) acc = fmaf(cat[k], w_cat[k * 32 + j], acc);
    zr[j] = elu_f(acc);
  }
  for (int j = 0; j < 32; ++j) {
    float acc = b_cont[j];
#pragma unroll
    for (int k = 0; k < 8; ++k) acc = fmaf(xc[k], w_cont[k * 32 + j], acc);
    zr[32 + j] = elu_f(acc);
  }
}

// ---------------------------------------------------------------------------
// WMMA f32 16x16x4 GEMM:  Y[nrows x (NT*16)] = act(X[nrows x 64] @ W + b)
// One wave computes MT row-tiles (MT*16 rows) x NT*16 cols. K=64 -> 16 steps.
// B fragments (weights) are loaded once per k-step and reused across the MT
// row-tiles; A/B loads for step k+1 are issued before the WMMAs of step k
// (software pipelining) so waits become partial instead of full drains.
// MODE 0: raw.  MODE 1: ELU(bias) then batchnorm affine.  MODE 2: ELU(bias).
// A-fragment: lane = half*16 + m holds A[row+m][4kk + 2*half + {0,1}].
// B-fragment: lane holds W[4kk + 2*half + {0,1}][ntile*16 + m].
// D/C: VGPR v, lane -> row (row0 + v + 8*half), col (ntile*16 + m).
// ---------------------------------------------------------------------------
template <int NT, int MT, int MODE>
__global__ __launch_bounds__(256) void gemm_wmma_kernel(
    const float* __restrict__ X, const float* __restrict__ W,
    const float* __restrict__ bias, const float* __restrict__ bn_g,
    const float* __restrict__ bn_b, const float* __restrict__ bn_m,
    const float* __restrict__ bn_v, float* __restrict__ Y, int nrows) {
  const int NOUT = NT * 16;
  int wave = (int)((blockIdx.x * blockDim.x + threadIdx.x) >> 5);
  int lane = (int)(threadIdx.x & 31);
  int m = lane & 15;
  int half = lane >> 4;
  int tile0 = wave * MT;
  if (tile0 * 16 >= nrows) return;  // wave-uniform: EXEC all-ones for WMMA

  // Per-tile A row pointers; clamp OOB rows (stores are guarded instead).
  const float* xrow[MT];
#pragma unroll
  for (int mt = 0; mt < MT; ++mt) {
    int rr = (tile0 + mt) * 16 + m;
    if (rr > nrows - 1) rr = nrows - 1;
    xrow[mt] = X + (size_t)rr * 64;
  }

  v8f acc[MT][NT] = {};
  v2f a[MT], b[NT];
  {
    const int kb = half * 2;
#pragma unroll
    for (int mt = 0; mt < MT; ++mt) a[mt] = *(const v2f*)(xrow[mt] + kb);
#pragma unroll
    for (int t = 0; t < NT; ++t) {
      b[t].x = W[(size_t)(kb + 0) * NOUT + t * 16 + m];
      b[t].y = W[(size_t)(kb + 1) * NOUT + t * 16 + m];
    }
  }
#pragma unroll
  for (int kk = 0; kk < 16; ++kk) {
    v2f an[MT], bn[NT];
    if (kk < 15) {
      const int kb2 = (kk + 1) * 4 + half * 2;
#pragma unroll
      for (int mt = 0; mt < MT; ++mt) an[mt] = *(const v2f*)(xrow[mt] + kb2);
#pragma unroll
      for (int t = 0; t < NT; ++t) {
        bn[t].x = W[(size_t)(kb2 + 0) * NOUT + t * 16 + m];
        bn[t].y = W[(size_t)(kb2 + 1) * NOUT + t * 16 + m];
      }
    }
#pragma unroll
    for (int mt = 0; mt < MT; ++mt) {
#pragma unroll
      for (int t = 0; t < NT; ++t) {
        acc[mt][t] = __builtin_amdgcn_wmma_f32_16x16x4_f32(
            false, a[mt], false, b[t], (short)0, acc[mt][t], false, false);
      }
    }
    if (kk < 15) {
#pragma unroll
      for (int mt = 0; mt < MT; ++mt) a[mt] = an[mt];
#pragma unroll
      for (int t = 0; t < NT; ++t) b[t] = bn[t];
    }
  }

  // Epilogue (per-column params are loop-invariant across mt).
#pragma unroll
  for (int t = 0; t < NT; ++t) {
    int c = t * 16 + m;
    float bb = 0.f, sc = 1.f, sh = 0.f;
    if constexpr (MODE == 1) {
      bb = bias[c];
      sc = bn_g[c] * rsqrtf(bn_v[c] + BN_EPS);
      sh = bn_b[c] - bn_m[c] * sc;
    }
    if constexpr (MODE == 2) { bb = bias[c]; }
#pragma unroll
    for (int mt = 0; mt < MT; ++mt) {
      int row0 = (tile0 + mt) * 16;
#pragma unroll
      for (int v = 0; v < 8; ++v) {
        int r = row0 + v + 8 * half;
        float val = acc[mt][t][v];
        if constexpr (MODE == 1) val = elu_f(val + bb) * sc + sh;
        if constexpr (MODE == 2) val = elu_f(val + bb);
        if (r < nrows) Y[(size_t)r * NOUT + c] = val;
      }
    }
  }
}

// ---------------------------------------------------------------------------
// GCN aggregation, CSR-by-destination. One wave32 per destination node:
// lane owns channels (lane, lane+32); per in-edge a coalesced 256B gather
// from h (resident in L2). No float atomics. Fused +bias, ReLU.
// ---------------------------------------------------------------------------
__global__ __launch_bounds__(256) void agg_kernel(
    const float* __restrict__ h, const float* __restrict__ dinv,
    const int* __restrict__ row_ptr, const int* __restrict__ col,
    const float* __restrict__ bias, float* __restrict__ out) {
  int node = (int)((blockIdx.x * blockDim.x + threadIdx.x) >> 5);
  int lane = (int)(threadIdx.x & 31);
  if (node >= N_NODES) return;
  float di = dinv[node];
  const float* hr = h + (size_t)node * HID;
  float a0 = di * di * hr[lane];
  float a1 = di * di * hr[lane + 32];
  int e0 = row_ptr[node], e1 = row_ptr[node + 1];
  for (int e = e0; e < e1; ++e) {
    int s = col[e];
    float w = di * dinv[s];
    const float* hs = h + (size_t)s * HID;
    a0 = fmaf(w, hs[lane], a0);
    a1 = fmaf(w, hs[lane + 32], a1);
  }
  a0 += bias[lane];
  a1 += bias[lane + 32];
  float* orow = out + (size_t)node * HID;
  orow[lane] = fmaxf(a0, 0.f);
  orow[lane + 32] = fmaxf(a1, 0.f);
}

// ---------------------------------------------------------------------------
// Final head: out = o1 @ out_w2 + b2 (32 -> 1 dot per node).
// ---------------------------------------------------------------------------
__global__ __launch_bounds__(256) void out2_kernel(const float* __restrict__ o1,
                                                   const float* __restrict__ w2,
                                                   const float* __restrict__ b2,
                                                   float* __restrict__ out) {
  int i = blockIdx.x * blockDim.x + threadIdx.x;
  if (i >= N_NODES) return;
  const float* r = o1 + (size_t)i * 32;
  float acc = b2[0];
#pragma unroll
  for (int k = 0; k < 32; ++k) acc = fmaf(r[k], w2[k], acc);
  out[i] = acc;
}

// ---------------------------------------------------------------------------
extern "C" void kernel_launch(void* const* d_in, const int* in_sizes, int n_in,
                              void* d_out, int out_size, void* d_ws,
                              size_t ws_size, hipStream_t stream) {
  (void)in_sizes; (void)n_in; (void)out_size; (void)ws_size;
  const float* x            = (const float*)d_in[0];
  const int*   ei           = (const int*)d_in[1];
  /* d_in[2] batch unused */
  const float* emb_charge_w = (const float*)d_in[3];
  const float* emb_pdgid_w  = (const float*)d_in[4];
  const float* emb_frompv_w = (const float*)d_in[5];
  const float* w_cont       = (const float*)d_in[6];
  const float* b_cont       = (const float*)d_in[7];
  const float* w_cat        = (const float*)d_in[8];
  const float* b_cat        = (const float*)d_in[9];
  const float* w_all        = (const float*)d_in[10];
  const float* b_all        = (const float*)d_in[11];
  const float* bn_g         = (const float*)d_in[12];
  const float* bn_b         = (const float*)d_in[13];
  const float* bn_m         = (const float*)d_in[14];
  const float* bn_v         = (const float*)d_in[15];
  const float* conv_w       = (const float*)d_in[16];
  const float* conv_b       = (const float*)d_in[17];
  const float* out_w1       = (const float*)d_in[18];
  const float* out_b1       = (const float*)d_in[19];
  const float* out_w2       = (const float*)d_in[20];
  const float* out_b2       = (const float*)d_in[21];
  float* out = (float*)d_out;

  char* ws = (char*)d_ws;
  size_t off = 0;
  auto alloc = [&](size_t bytes) -> char* {
    char* p = ws + off;
    off += (bytes + 255) & ~(size_t)255;
    return p;
  };
  float* buf0   = (float*)alloc((size_t)N_NODES * HID * sizeof(float));
  float* buf1   = (float*)alloc((size_t)N_NODES * HID * sizeof(float));
  float* buf2   = (float*)alloc((size_t)N_NODES * HID * sizeof(float));
  int* deg      = (int*)alloc((size_t)N_NODES * sizeof(int));
  float* dinv   = (float*)alloc((size_t)N_NODES * sizeof(float));
  int* row_ptr  = (int*)alloc((size_t)(N_NODES + 1) * sizeof(int));
  int* nxt      = (int*)alloc((size_t)N_NODES * sizeof(int));
  int* col      = (int*)alloc((size_t)N_EDGES * sizeof(int));

  hipMemsetAsync(deg, 0, (size_t)N_NODES * sizeof(int), stream);

  const int EB = (N_EDGES + 255) / 256;
  const int NB = (N_NODES + 255) / 256;
  const int MT = 4;  // row tiles per wave in the WMMA GEMM
  const int TILES = (N_NODES + 15) / 16;
  const int GWAVES = (TILES + MT - 1) / MT;
  const int GB = (GWAVES * 32 + 255) / 256;
  const int AB = (N_NODES * 32 + 255) / 256;  // 1 wave per node

  deg_kernel<<<EB, 256, 0, stream>>>(ei, deg);
  dinv_kernel<<<NB, 256, 0, stream>>>(deg, dinv);
  scan_kernel<<<1, 1024, 0, stream>>>(deg, row_ptr, nxt, N_NODES);
  fill_kernel<<<EB, 256, 0, stream>>>(ei, nxt, col);

  frontend_kernel<<<NB, 256, 0, stream>>>(x, emb_charge_w, emb_pdgid_w,
                                          emb_frompv_w, w_cont, b_cont, w_cat,
                                          b_cat, buf0);

  // emb = BN(elu(z @ w_all + b_all))
  gemm_wmma_kernel<4, 4, 1><<<GB, 256, 0, stream>>>(
      buf0, w_all, b_all, bn_g, bn_b, bn_m, bn_v, buf1, N_NODES);
  // conv layer 0
  gemm_wmma_kernel<4, 4, 0><<<GB, 256, 0, stream>>>(
      buf1, conv_w, nullptr, nullptr, nullptr, nullptr, nullptr, buf2,
      N_NODES);
  agg_kernel<<<AB, 256, 0, stream>>>(buf2, dinv, row_ptr, col, conv_b, buf0);
  // conv layer 1
  gemm_wmma_kernel<4, 4, 0><<<GB, 256, 0, stream>>>(
      buf0, conv_w + HID * HID, nullptr, nullptr, nullptr, nullptr, nullptr,
      buf2, N_NODES);
  agg_kernel<<<AB, 256, 0, stream>>>(buf2, dinv, row_ptr, col, conv_b + HID,
                                     buf1);
  // head: o1 = elu(emb @ out_w1 + out_b1)
  gemm_wmma_kernel<2, 4, 2><<<GB, 256, 0, stream>>>(
      buf1, out_w1, out_b1, nullptr, nullptr, nullptr, nullptr, buf0, N_NODES);
  out2_kernel<<<NB, 256, 0, stream>>>(buf0, out_w2, out_b2, out);
}